// DA_conv_67765993996497
// MI455X (gfx1250) — compile-verified
//
#include <hip/hip_runtime.h>
#include <math.h>

#define NEG_SLOPE 0.1f
#define Bn   16
#define Cn   64
#define Hn   128
#define Wn   128
#define DEGn 512
#define REDn 8
#define CK2  576      // C * 3 * 3
#define TILE_P 64
#define CSTRIDE 68    // bank = (4p + c) mod 64 -> conflict-free b64 B-fragment loads

typedef float v2f __attribute__((ext_vector_type(2)));
typedef float v8f __attribute__((ext_vector_type(8)));

__device__ __forceinline__ float lrelu(float x) { return x >= 0.f ? x : NEG_SLOPE * x; }

// ---------------------------------------------------------------------------
// Kernel 1: per-sample tiny MLPs. One block per batch element.
//   dvec = mean(deg[b], HW)                       [512]
//   kern = lrelu(dvec@Ws^T @ Wk1^T) @ Wk2^T       [576]  -> ws
//   att  = sigmoid(lrelu(dvec@Wac^T @ Wdu1^T) @ Wdu2^T)  -> ws
// ---------------------------------------------------------------------------
__global__ __launch_bounds__(256) void prep_kernel(
    const float* __restrict__ deg,  const float* __restrict__ W_size,
    const float* __restrict__ W_k1, const float* __restrict__ W_k2,
    const float* __restrict__ W_ac, const float* __restrict__ W_du1,
    const float* __restrict__ W_du2,
    float* __restrict__ kern_ws, float* __restrict__ att_ws)
{
    __shared__ float sdvec[DEGn];
    __shared__ float sf[Cn], sfa[Cn];
    __shared__ float sg[REDn], sgu[REDn];
    const int b = blockIdx.x, t = threadIdx.x;

    for (int d = t; d < DEGn; d += 256) {
        const float* p = deg + ((size_t)b * DEGn + d) * 64;
        float s = 0.f;
        for (int i = 0; i < 64; ++i) s += p[i];
        sdvec[d] = s * (1.f / 64.f);
    }
    __syncthreads();

    if (t < Cn) {                       // f = dvec @ W_size^T
        const float* wr = W_size + (size_t)t * DEGn;
        float s = 0.f;
        for (int d = 0; d < DEGn; ++d) s += wr[d] * sdvec[d];
        sf[t] = s;
    } else if (t < 2 * Cn) {            // fa = dvec @ W_ac^T
        const int c = t - Cn;
        const float* wr = W_ac + (size_t)c * DEGn;
        float s = 0.f;
        for (int d = 0; d < DEGn; ++d) s += wr[d] * sdvec[d];
        sfa[c] = s;
    }
    __syncthreads();

    if (t < REDn) {                     // g = lrelu(f @ W_k1^T)
        const float* wr = W_k1 + t * Cn;
        float s = 0.f;
        for (int c = 0; c < Cn; ++c) s += wr[c] * sf[c];
        sg[t] = lrelu(s);
    } else if (t < 2 * REDn) {          // gu = lrelu(fa @ W_du1^T)
        const int r = t - REDn;
        const float* wr = W_du1 + r * Cn;
        float s = 0.f;
        for (int c = 0; c < Cn; ++c) s += wr[c] * sfa[c];
        sgu[r] = lrelu(s);
    }
    __syncthreads();

    for (int j = t; j < CK2; j += 256) { // kern = g @ W_k2^T
        const float* wr = W_k2 + j * REDn;
        float s = 0.f;
        for (int r = 0; r < REDn; ++r) s += wr[r] * sg[r];
        kern_ws[b * CK2 + j] = s;
    }
    if (t < Cn) {                        // att = sigmoid(gu @ W_du2^T)
        const float* wr = W_du2 + t * REDn;
        float s = 0.f;
        for (int r = 0; r < REDn; ++r) s += wr[r] * sgu[r];
        att_ws[b * Cn + t] = 1.f / (1.f + expf(-s));
    }
}

// ---------------------------------------------------------------------------
// Kernel 2: fused depthwise-3x3 + lrelu + 1x1-conv(WMMA f32) + bias + residual
// Block = (b, row h, 64-col tile). 256 threads = 8 waves.
// Mid buffer stored TRANSPOSED: sMt[pos][channel] so each WMMA B-fragment
// ({B[c][p], B[c+1][p]}) is one aligned ds_load_b64, no register marshalling.
// ---------------------------------------------------------------------------
__global__ __launch_bounds__(256) void fused_kernel(
    const float* __restrict__ feat, const float* __restrict__ W_conv,
    const float* __restrict__ b_conv,
    const float* __restrict__ kern_ws, const float* __restrict__ att_ws,
    float* __restrict__ out)
{
    __shared__ float sMt[TILE_P][CSTRIDE]; // lrelu(depthwise) : [pos][channel]
    __shared__ float sK[Cn][9];
    __shared__ float sBias[Cn];
    __shared__ float sAtt[Cn];

    const int b  = blockIdx.z;
    const int h  = blockIdx.y;
    const int w0 = blockIdx.x * TILE_P;
    const int t  = threadIdx.x;

    if (t < Cn) { sBias[t] = b_conv[t]; sAtt[t] = att_ws[b * Cn + t]; }
    for (int j = t; j < Cn * 9; j += 256) (&sK[0][0])[j] = kern_ws[b * CK2 + j];
    __syncthreads();

    // ---- depthwise 3x3 + lrelu (zero padding), 16 (c,p) entries per thread
    for (int i = 0; i < 16; ++i) {
        const int idx = t + i * 256;      // 0..4095
        const int c = idx >> 6, p = idx & 63;
        const int w = w0 + p;
        const float* kc = &sK[c][0];
        float acc = 0.f;
        #pragma unroll
        for (int dr = 0; dr < 3; ++dr) {
            const int hh = h + dr - 1;
            if (hh < 0 || hh >= Hn) continue;           // uniform per block
            const float* rowp = feat + (((size_t)(b * Cn + c)) * Hn + hh) * Wn;
            #pragma unroll
            for (int dc = 0; dc < 3; ++dc) {
                const int ww = w + dc - 1;
                const float v = (ww >= 0 && ww < Wn) ? rowp[ww] : 0.f;
                acc = fmaf(kc[dr * 3 + dc], v, acc);
            }
        }
        sMt[p][c] = lrelu(acc);
    }
    __syncthreads();

    // ---- 1x1 conv as GEMM: D[o,p] = sum_c Wc[o,c] * sMt[p,c]
    // wave -> 16 out-channels x 32 positions (two 16x16 accumulators)
    const int wave = t >> 5;
    const int lane = t & 31;
    const int half = lane >> 4;      // 0: lanes 0-15, 1: lanes 16-31
    const int ln   = lane & 15;
    const int o0   = (wave >> 1) * 16;
    const int p0   = (wave & 1) * 32;

    // A fragments (16x4 f32): lane<16 holds K=+0/+1, lane>=16 holds K=+2/+3
    v2f afrag[16];
    const float* wrow = W_conv + (size_t)(o0 + ln) * Cn + 2 * half;
    #pragma unroll
    for (int kk = 0; kk < 16; ++kk)
        afrag[kk] = *(const v2f*)(wrow + 4 * kk);       // 8B-aligned global_load_b64

    v8f acc0 = {}; v8f acc1 = {};
    #pragma unroll
    for (int kk = 0; kk < 16; ++kk) {
        const int c0 = 4 * kk + 2 * half;  // B 4x16: VGPR0 rows K=0/2, VGPR1 rows K=1/3
        const v2f b0 = *(const v2f*)&sMt[p0 + ln][c0];       // one ds_load_b64
        const v2f b1 = *(const v2f*)&sMt[p0 + 16 + ln][c0];  // one ds_load_b64
        acc0 = __builtin_amdgcn_wmma_f32_16x16x4_f32(false, afrag[kk], false, b0,
                                                     (short)0, acc0, false, false);
        acc1 = __builtin_amdgcn_wmma_f32_16x16x4_f32(false, afrag[kk], false, b1,
                                                     (short)0, acc1, false, false);
    }

    // ---- epilogue: + bias + feat*att, store
    #pragma unroll
    for (int r = 0; r < 8; ++r) {
        const int o = o0 + r + 8 * half;   // C/D: VGPR r -> rows r (lo half) / r+8 (hi half)
        const size_t base = (((size_t)(b * Cn + o)) * Hn + h) * Wn + w0;
        const float fa0 = feat[base + p0 + ln];
        const float fa1 = feat[base + p0 + 16 + ln];
        out[base + p0 + ln]      = acc0[r] + sBias[o] + fa0 * sAtt[o];
        out[base + p0 + 16 + ln] = acc1[r] + sBias[o] + fa1 * sAtt[o];
    }
}

extern "C" void kernel_launch(void* const* d_in, const int* in_sizes, int n_in,
                              void* d_out, int out_size, void* d_ws, size_t ws_size,
                              hipStream_t stream)
{
    const float* feat   = (const float*)d_in[0];
    const float* deg    = (const float*)d_in[1];
    const float* W_size = (const float*)d_in[2];
    const float* W_k1   = (const float*)d_in[3];
    const float* W_k2   = (const float*)d_in[4];
    const float* W_conv = (const float*)d_in[5];
    const float* b_conv = (const float*)d_in[6];
    const float* W_ac   = (const float*)d_in[7];
    const float* W_du1  = (const float*)d_in[8];
    const float* W_du2  = (const float*)d_in[9];
    float* out = (float*)d_out;

    float* kern_ws = (float*)d_ws;             // [B,576]
    float* att_ws  = kern_ws + Bn * CK2;       // [B,64]

    prep_kernel<<<dim3(Bn), dim3(256), 0, stream>>>(
        deg, W_size, W_k1, W_k2, W_ac, W_du1, W_du2, kern_ws, att_ws);

    fused_kernel<<<dim3(Wn / TILE_P, Hn, Bn), dim3(256), 0, stream>>>(
        feat, W_conv, b_conv, kern_ws, att_ws, out);
}